// FusedRiemannEngine_53584011985646
// MI455X (gfx1250) — compile-verified
//
// MI455X (gfx1250) fused kernel: batched 1024-pt FFT (bandwidth-bound, ~256 MB
// -> ~11us roofline @ 23.3 TB/s) + Riemann-Siegel Z cosine sum (trivial).
// CDNA5 paths: global_load_async_to_lds_b128 / global_store_async_from_lds_b128
// (+ s_wait_asynccnt), V_WMMA_F32_16X16X4_F32 for the first four DIT stages.
// Twiddles use native __sincosf (args < pi) -- no Payne-Hanek bloat; only the
// zeta kernel keeps precise cosf (phase ~5e5 rad, must match reference fp32).
#include <hip/hip_runtime.h>
#include <cstdint>

#define PI_F      3.14159265358979323846f
#define PI_OVER8F 0.39269908169872415481f
#define TWO_PI_F  6.28318530717958647692f
#define LOG2PI_F  1.83787706640934548356f

typedef float v2f __attribute__((ext_vector_type(2)));
typedef float v8f __attribute__((ext_vector_type(8)));

#if defined(__gfx1250__) && __has_builtin(__builtin_amdgcn_wmma_f32_16x16x4_f32)
#define HAVE_WMMA_F32 1
#endif
#if defined(__gfx1250__)
#define HAVE_ASYNC_LDS 1
#endif

#if defined(HAVE_WMMA_F32)
__device__ __forceinline__ v8f wmma_f32_16x16x4(v2f a, v2f b, v8f c) {
  // (neg_a, A, neg_b, B, c_mod, C, reuse_a, reuse_b)
  return __builtin_amdgcn_wmma_f32_16x16x4_f32(false, a, false, b, (short)0, c,
                                               false, false);
}
#endif

#if defined(HAVE_ASYNC_LDS)
// One wave instruction moves 32 lanes x 16B memory<->LDS, tracked by ASYNCcnt.
#define ASYNC_LOAD_B128(ldsp, gp)                                              \
  {                                                                            \
    unsigned _d = (unsigned)(uintptr_t)(ldsp);                                 \
    unsigned long long _a = (unsigned long long)(uintptr_t)(gp);               \
    asm volatile("global_load_async_to_lds_b128 %0, %1, off" ::"v"(_d),        \
                 "v"(_a)                                                       \
                 : "memory");                                                  \
  }
#define ASYNC_STORE_B128(gp, ldsp)                                             \
  {                                                                            \
    unsigned _s = (unsigned)(uintptr_t)(ldsp);                                 \
    unsigned long long _a = (unsigned long long)(uintptr_t)(gp);               \
    asm volatile("global_store_async_from_lds_b128 %0, %1, off" ::"v"(_a),     \
                 "v"(_s)                                                       \
                 : "memory");                                                  \
  }
#if __has_builtin(__builtin_amdgcn_s_wait_asynccnt)
#define WAIT_ASYNC0() __builtin_amdgcn_s_wait_asynccnt(0)
#else
#define WAIT_ASYNC0() asm volatile("s_wait_asynccnt 0" ::: "memory")
#endif
#endif

// ---------------------------------------------------------------------------
// FFT: one block (128 threads = 4 wave32) per 1024-point row.
// Plan: async-load row -> LDS; stages 0..3 fused as a 16x16 DFT matmul via
// fp32 WMMA (O[m][c] = sum_j e^{-2pi i m br4(j)/16} x[br10(16c+j)]);
// stages 4..9 as LDS butterflies; async-store LDS -> row.
// ---------------------------------------------------------------------------
__global__ __launch_bounds__(128) void fft1024_kernel(
    const float* __restrict__ xre, const float* __restrict__ xim,
    float* __restrict__ yre, float* __restrict__ yim) {
  __shared__ float lxre[1024];
  __shared__ float lxim[1024];
  __shared__ float lyre[1024];
  __shared__ float lyim[1024];

  const int tid = threadIdx.x;
  const size_t row = (size_t)blockIdx.x * 1024u;
  const int i0 = tid * 4;        // floats 0..511, 16B per lane, coalesced
  const int i1 = 512 + tid * 4;  // floats 512..1023

#if defined(HAVE_ASYNC_LDS)
  ASYNC_LOAD_B128(&lxre[i0], xre + row + i0);
  ASYNC_LOAD_B128(&lxre[i1], xre + row + i1);
  ASYNC_LOAD_B128(&lxim[i0], xim + row + i0);
  ASYNC_LOAD_B128(&lxim[i1], xim + row + i1);
  WAIT_ASYNC0();
#else
  for (int i = tid; i < 1024; i += 128) {
    lxre[i] = xre[row + i];
    lxim[i] = xim[row + i];
  }
#endif
  __syncthreads();

#if defined(HAVE_WMMA_F32)
  // First four DIT stages as one complex 16x16 matmul per 16-column tile.
  // Wave w (0..3) owns output columns c = 16w .. 16w+15.
  {
    const int lane = tid & 31;
    const int w = tid >> 5;
    const int mrow = lane & 15;   // A-matrix row m / B,C column n
    const int hb = lane >> 4;     // lane half -> K-chunk offset (32-bit layout)
    const int jb = hb * 2;
    const int c = 16 * w + mrow;  // this lane's B/C column

    v8f accRe = {};
    v8f accIm = {};
#pragma unroll
    for (int k = 0; k < 4; ++k) {
      const int j0 = 4 * k + jb;
      const int j1 = j0 + 1;
      // A fragments: theta = (pi/8)*((m*br4(j)) mod 16); tiny args -> native.
      const int t0 = (mrow * (int)(__brev((unsigned)j0) >> 28)) & 15;
      const int t1 = (mrow * (int)(__brev((unsigned)j1) >> 28)) & 15;
      float s0, c0, s1, c1;
      __sincosf(PI_OVER8F * (float)t0, &s0, &c0);
      __sincosf(PI_OVER8F * (float)t1, &s1, &c1);
      v2f aC = {c0, c1};    // Re(M) = cos
      v2f aS = {s0, s1};    // -Im(M) = sin
      v2f aSn = {-s0, -s1};
      // B fragments: rows j of G, G[j][c] = x[br10(16c + j)]
      const unsigned q0 = __brev((unsigned)(16 * c + j0)) >> 22;
      const unsigned q1 = __brev((unsigned)(16 * c + j1)) >> 22;
      v2f bRe = {lxre[q0], lxre[q1]};
      v2f bIm = {lxim[q0], lxim[q1]};
      // Ore = cos*Gre + sin*Gim ; Oim = cos*Gim - sin*Gre
      accRe = wmma_f32_16x16x4(aC, bRe, accRe);
      accRe = wmma_f32_16x16x4(aS, bIm, accRe);
      accIm = wmma_f32_16x16x4(aC, bIm, accIm);
      accIm = wmma_f32_16x16x4(aSn, bRe, accIm);
    }
    // C/D layout: vgpr r -> M=r (lanes 0-15) / M=r+8 (lanes 16-31), N=lane&15.
#pragma unroll
    for (int r = 0; r < 8; ++r) {
      const int mm = r + (hb ? 8 : 0);
      lyre[16 * c + mm] = accRe[r];
      lyim[16 * c + mm] = accIm[r];
    }
  }
  const int START_STAGE = 4;
#else
  // Fallback: bit-reversal copy, then run all 10 stages as butterflies.
  for (int p = tid; p < 1024; p += 128) {
    const unsigned q = __brev((unsigned)p) >> 22;
    lyre[p] = lxre[q];
    lyim[p] = lxim[q];
  }
  const int START_STAGE = 0;
#endif

  // Remaining radix-2 stages: 512 butterflies/stage over 128 threads.
  // Twiddle angle pi*j/half < pi -> native __sincosf (no range reduction).
  for (int s = START_STAGE; s < 10; ++s) {
    __syncthreads();
    const int half = 1 << s;
    const float astep = PI_F / (float)half;
#pragma unroll 4
    for (int bi = tid; bi < 512; bi += 128) {
      const int j = bi & (half - 1);
      const int p0 = ((bi >> s) << (s + 1)) + j;
      const int p1 = p0 + half;
      float sv, cv;
      __sincosf(astep * (float)j, &sv, &cv);  // w = cv - i*sv
      const float ur = lyre[p0], ui = lyim[p0];
      const float vr = lyre[p1], vi = lyim[p1];
      const float tr = vr * cv + vi * sv;   // vr*wr - vi*wi, wi = -sv
      const float ti = vi * cv - vr * sv;   // vr*wi + vi*wr
      lyre[p0] = ur + tr;
      lyim[p0] = ui + ti;
      lyre[p1] = ur - tr;
      lyim[p1] = ui - ti;
    }
  }
  __syncthreads();

#if defined(HAVE_ASYNC_LDS)
  // Stream results LDS -> global without a VGPR round trip.
  ASYNC_STORE_B128(yre + row + i0, &lyre[i0]);
  ASYNC_STORE_B128(yre + row + i1, &lyre[i1]);
  ASYNC_STORE_B128(yim + row + i0, &lyim[i0]);
  ASYNC_STORE_B128(yim + row + i1, &lyim[i1]);
  WAIT_ASYNC0();
#else
  for (int i = tid; i < 1024; i += 128) {
    yre[row + i] = lyre[i];
    yim[row + i] = lyim[i];
  }
#endif
}

// ---------------------------------------------------------------------------
// Riemann-Siegel Z(t): Z = 2 * sum_{n=1..nt} cos(t*ln n - theta)/sqrt(n),
// nt = clip(int(sqrt(t/2pi)), 10, 1024). LDS tables for ln n, rsqrt n.
// Phase is ~5e5 rad where 1 ulp ~ 0.06 rad: keep precise cosf and mirror the
// reference's exact fp32 op order (mul, then sub; no fma).
// ---------------------------------------------------------------------------
__global__ __launch_bounds__(256) void zeta_kernel(const float* __restrict__ t,
                                                   float* __restrict__ Z,
                                                   int n) {
  __shared__ float slog[1024];
  __shared__ float srs[1024];
  for (int i = threadIdx.x; i < 1024; i += 256) {
    const float fn = (float)(i + 1);
    slog[i] = logf(fn);
    srs[i] = rsqrtf(fn);
  }
  __syncthreads();

  const int gid = blockIdx.x * 256 + threadIdx.x;
  if (gid >= n) return;
  const float tv = t[gid];
  const float th = 0.5f * tv * (logf(tv) - LOG2PI_F) - 0.5f * tv -
                   PI_F * 0.125f + 1.0f / (48.0f * tv);
  int nt = (int)sqrtf(tv * (1.0f / TWO_PI_F));
  nt = nt < 10 ? 10 : (nt > 1024 ? 1024 : nt);
  float acc = 0.0f;
  for (int i = 0; i < nt; ++i) {
    const float phase = tv * slog[i] - th;  // same rounding seq as reference
    acc += srs[i] * cosf(phase);
  }
  Z[gid] = 2.0f * acc;
}

// ---------------------------------------------------------------------------
extern "C" void kernel_launch(void* const* d_in, const int* in_sizes, int n_in,
                              void* d_out, int out_size, void* d_ws,
                              size_t ws_size, hipStream_t stream) {
  (void)n_in;
  (void)d_ws;
  (void)ws_size;
  (void)out_size;
  const float* xre = (const float*)d_in[0];
  const float* xim = (const float*)d_in[1];
  const float* t = (const float*)d_in[2];

  const int B = in_sizes[0] / 1024;  // 16384 rows
  const int nT = in_sizes[2];        // 262144 points

  float* yre = (float*)d_out;
  float* yim = yre + (size_t)B * 1024u;
  float* Z = yre + 2u * (size_t)B * 1024u;

  fft1024_kernel<<<B, 128, 0, stream>>>(xre, xim, yre, yim);
  zeta_kernel<<<(nT + 255) / 256, 256, 0, stream>>>(t, Z, nT);
}